// HF_80573586473140
// MI455X (gfx1250) — compile-verified
//
#include <hip/hip_runtime.h>
#include <stdint.h>

// Householder reflection: out = z - 2 * v * (v.z) / (v.v), rowwise over [B, L].
// Memory-bound (0.58 FLOP/byte): strategy is single HBM read of v and z via
// CDNA5 async global->LDS B128 copies, LDS-resident second pass, NT B128 stores.

typedef __attribute__((ext_vector_type(4))) float v4f;

constexpr int L          = 2048;       // row length (fixed by reference)
constexpr int NV4        = L / 4;      // 512 float4 per row
constexpr int TPB        = 256;        // 8 wave32 waves
constexpr int V4_PER_THR = NV4 / TPB;  // 2 float4 per thread per array
constexpr int NWAVES     = TPB / 32;

__global__ __launch_bounds__(TPB) void hh_reflect_kernel(
    const float* __restrict__ v,
    const float* __restrict__ z,
    float* __restrict__ out)
{
    __shared__ v4f lds_v[NV4];             // 8 KiB
    __shared__ v4f lds_z[NV4];             // 8 KiB
    __shared__ float red_vz[NWAVES];
    __shared__ float red_vv[NWAVES];

    const int row = blockIdx.x;
    const int tid = threadIdx.x;

    const float* __restrict__ vrow = v + (size_t)row * L;
    const float* __restrict__ zrow = z + (size_t)row * L;

    // ---- Phase 1: async DMA row of v and z into LDS (B128 per lane) ----
    // ASYNCcnt-tracked; LDS byte address in a VGPR, 64-bit global addr in a
    // VGPR pair (GV mode). Low 32 bits of a flat LDS pointer are the logical
    // LDS byte address.
#pragma unroll
    for (int j = 0; j < V4_PER_THR; ++j) {
        const int i = tid + j * TPB;                   // float4 index in row
        uint32_t lv = (uint32_t)(uintptr_t)(&lds_v[i]);
        uint32_t lz = (uint32_t)(uintptr_t)(&lds_z[i]);
        uint64_t gv = (uint64_t)(uintptr_t)(vrow + 4 * i);
        uint64_t gz = (uint64_t)(uintptr_t)(zrow + 4 * i);
        asm volatile("global_load_async_to_lds_b128 %0, %1, off"
                     :: "v"(lv), "v"(gv) : "memory");
        asm volatile("global_load_async_to_lds_b128 %0, %1, off"
                     :: "v"(lz), "v"(gz) : "memory");
    }
    asm volatile("s_wait_asynccnt 0" ::: "memory");    // LDS writes landed
    __syncthreads();                                   // visible to all waves

    // ---- Phase 2: partial dot products from LDS (ds_load_b128) ----
    float vz = 0.0f, vv = 0.0f;
#pragma unroll
    for (int j = 0; j < V4_PER_THR; ++j) {
        const int i = tid + j * TPB;
        const v4f a = lds_v[i];
        const v4f b = lds_z[i];
        vz += a.x * b.x + a.y * b.y + a.z * b.z + a.w * b.w;
        vv += a.x * a.x + a.y * a.y + a.z * a.z + a.w * a.w;
    }

    // wave32 butterfly reduction (gfx1250 is wave32-only)
#pragma unroll
    for (int m = 16; m > 0; m >>= 1) {
        vz += __shfl_xor(vz, m, 32);
        vv += __shfl_xor(vv, m, 32);
    }
    const int wave = tid >> 5;
    const int lane = tid & 31;
    if (lane == 0) { red_vz[wave] = vz; red_vv[wave] = vv; }
    __syncthreads();

    // deterministic fixed-order final sum (every thread, same order)
    float tvz = 0.0f, tvv = 0.0f;
#pragma unroll
    for (int w = 0; w < NWAVES; ++w) { tvz += red_vz[w]; tvv += red_vv[w]; }
    const float scale = 2.0f * tvz / tvv;

    // ---- Phase 3: out = z - scale * v, NT B128 streaming stores ----
    v4f* __restrict__ orow = (v4f*)(out + (size_t)row * L);
#pragma unroll
    for (int j = 0; j < V4_PER_THR; ++j) {
        const int i = tid + j * TPB;
        const v4f a = lds_v[i];
        const v4f b = lds_z[i];
        v4f r;
        r.x = b.x - scale * a.x;
        r.y = b.y - scale * a.y;
        r.z = b.z - scale * a.z;
        r.w = b.w - scale * a.w;
        __builtin_nontemporal_store(r, &orow[i]);
    }
}

extern "C" void kernel_launch(void* const* d_in, const int* in_sizes, int n_in,
                              void* d_out, int out_size, void* d_ws, size_t ws_size,
                              hipStream_t stream) {
    const float* v = (const float*)d_in[0];
    const float* z = (const float*)d_in[1];
    float* out     = (float*)d_out;

    const int B = in_sizes[0] / L;   // 8192 rows
    hh_reflect_kernel<<<B, TPB, 0, stream>>>(v, z, out);
}